// Inference_15015205667415
// MI455X (gfx1250) — compile-verified
//
#include <hip/hip_runtime.h>
#include <stdint.h>

#define WAVES_PER_BLOCK 8
#define BLOCK (WAVES_PER_BLOCK * 32)
#define PIX_PER_WAVE 16
#define KCH 128
#define SUPPORT 3
#define DEPTH 8              // async pipeline depth (ring of LDS slots per wave)

// Async copy of 16B per lane (512B per wave) from global memory into LDS.
// CDNA5 GLOBAL_LOAD_ASYNC_TO_LDS_B128: tracked by ASYNCcnt, no VGPR data return.
__device__ __forceinline__ void async_copy16(uint32_t lds_byte_off, const void* gaddr) {
    asm volatile("global_load_async_to_lds_b128 %0, %1, off"
                 :: "v"(lds_byte_off), "v"(gaddr)
                 : "memory");
}
// Wait until at most 7 async ops outstanding => oldest of 8 has completed.
__device__ __forceinline__ void wait_async7() {
    asm volatile("s_wait_asynccnt 0x7" ::: "memory");
}

__global__ __launch_bounds__(BLOCK)
void subexp_kernel(const float* __restrict__ beliefs, float* __restrict__ out, int nPix) {
    // 8 waves * 8 ring slots * 32 lanes * 16B = 32 KB LDS per block.
    __shared__ float4 stage[WAVES_PER_BLOCK][DEPTH][32];

    const int lane = threadIdx.x & 31;
    const int wave = threadIdx.x >> 5;
    const long long waveGid = (long long)blockIdx.x * WAVES_PER_BLOCK + wave;
    const long long p0 = waveGid * PIX_PER_WAVE;
    if (p0 >= nPix) return;
    const int npix = (int)((nPix - p0 < PIX_PER_WAVE) ? (nPix - p0) : PIX_PER_WAVE);

    // Each lane owns channels [4*lane, 4*lane+3] of its wave's pixel.
    const float4* gbase = (const float4*)beliefs + p0 * (KCH / 4) + lane;
    // Low 32 bits of a generic LDS pointer == byte offset into LDS.
    const uint32_t ldsBase = (uint32_t)(uintptr_t)&stage[wave][0][lane];
    // Ring slot stride: 32 lanes * 16B = 512 bytes.

    const int   ik0 = lane * 4;
    const float fk0 = (float)ik0;

    // Prologue: fill DEPTH-1 ring slots (clamped pixel index for short tails).
    #pragma unroll
    for (int j = 0; j < DEPTH - 1; ++j) {
        const int pj = (j < npix) ? j : (npix - 1);
        async_copy16(ldsBase + (uint32_t)j * 512u,
                     (const void*)(gbase + (long long)pj * (KCH / 4)));
    }

    for (int i = 0; i < npix; ++i) {
        // Issue prefetch for pixel i+7 (clamped: tail re-loads last pixel, harmless),
        // keeping 8 async ops in flight; then wait so pixel i's copy has landed.
        const int pn0 = i + (DEPTH - 1);
        const int pn  = (pn0 < npix) ? pn0 : (npix - 1);
        async_copy16(ldsBase + (uint32_t)(pn0 & (DEPTH - 1)) * 512u,
                     (const void*)(gbase + (long long)pn * (KCH / 4)));
        wait_async7();

        const float4 v = stage[wave][i & (DEPTH - 1)][lane];

        // ---- 1) wave-wide max over K=128 ----
        float m = fmaxf(fmaxf(v.x, v.y), fmaxf(v.z, v.w));
        #pragma unroll
        for (int o = 16; o >= 1; o >>= 1) m = fmaxf(m, __shfl_xor(m, o, 32));

        // ---- 2) first-occurrence argmax (lowest index among maxima) ----
        int idx = 0x7FFFFFFF;
        idx = (v.w == m) ? (ik0 + 3) : idx;
        idx = (v.z == m) ? (ik0 + 2) : idx;
        idx = (v.y == m) ? (ik0 + 1) : idx;
        idx = (v.x == m) ? (ik0 + 0) : idx;
        #pragma unroll
        for (int o = 16; o >= 1; o >>= 1) idx = min(idx, __shfl_xor(idx, o, 32));
        const int disp = __builtin_amdgcn_readfirstlane(idx);

        // ---- 3) clamped-window weighted sums (multiplicity trick, no gather) ----
        int lo = disp - SUPPORT; lo = (lo < 0) ? 0 : lo;
        int hi = disp + SUPPORT; hi = (hi > KCH - 1) ? (KCH - 1) : hi;
        const unsigned span = (unsigned)(hi - lo);

        const float s0 = ((unsigned)(ik0 + 0 - lo) <= span) ? v.x : 0.0f;
        const float s1 = ((unsigned)(ik0 + 1 - lo) <= span) ? v.y : 0.0f;
        const float s2 = ((unsigned)(ik0 + 2 - lo) <= span) ? v.z : 0.0f;
        const float s3 = ((unsigned)(ik0 + 3 - lo) <= span) ? v.w : 0.0f;

        float sw  = (s0 + s1) + (s2 + s3);
        float swk = fmaf(s3, fk0 + 3.0f,
                    fmaf(s2, fk0 + 2.0f,
                    fmaf(s1, fk0 + 1.0f, s0 * fk0)));

        // Edge multiplicity from clip(): duplicated index 0 / 127.
        // k=0 extra adds only to the denominator (weight * 0 == 0).
        const float e0 = (ik0 == 0)
                       ? (float)((disp < SUPPORT) ? (SUPPORT - disp) : 0) : 0.0f;
        sw = fmaf(e0, v.x, sw);
        const float e1 = (ik0 == KCH - 4)
                       ? (float)((disp > KCH - 1 - SUPPORT) ? (disp - (KCH - 1 - SUPPORT)) : 0)
                       : 0.0f;
        sw  = fmaf(e1, v.w, sw);
        swk = fmaf(e1 * v.w, (float)(KCH - 1), swk);

        #pragma unroll
        for (int o = 16; o >= 1; o >>= 1) {
            sw  += __shfl_xor(sw,  o, 32);
            swk += __shfl_xor(swk, o, 32);
        }

        if (lane == 0) out[p0 + i] = swk / sw;
    }
}

extern "C" void kernel_launch(void* const* d_in, const int* in_sizes, int n_in,
                              void* d_out, int out_size, void* d_ws, size_t ws_size,
                              hipStream_t stream) {
    const float* beliefs = (const float*)d_in[0];
    float* out = (float*)d_out;
    const int nPix = out_size;  // N*H*W = 983040 ; in_sizes[0]/out_size == 128 == KCH

    const int wavesNeeded = (nPix + PIX_PER_WAVE - 1) / PIX_PER_WAVE;
    const int blocks = (wavesNeeded + WAVES_PER_BLOCK - 1) / WAVES_PER_BLOCK;
    subexp_kernel<<<blocks, BLOCK, 0, stream>>>(beliefs, out, nPix);
}